// CTRNN_13640816132645
// MI455X (gfx1250) — compile-verified
//
#include <hip/hip_runtime.h>

typedef float v2f __attribute__((ext_vector_type(2)));
typedef float v8f __attribute__((ext_vector_type(8)));

constexpr int kT = 512;    // seq_len
constexpr int kB = 128;    // batch
constexpr int kI = 256;    // input_size
constexpr int kH = 1024;   // hidden_size
constexpr float kAlpha = 0.1f;

constexpr int kNWG     = 128;              // 8 m-strips x 16 n-groups, 1 WG per WGP
constexpr int kBK      = 128;              // K-block staged per LDS buffer
constexpr int kAStride = kBK + 4;          // 132: banks (4r+2c) distinct, rows 16B-aligned
constexpr int kBStride = kH + 4;           // 1028: banks (4r+2c) distinct, rows 16B-aligned
constexpr int kWSlice  = 16 * kBStride;    // floats per wave's resident Wh slice
constexpr int kSBFloats = 4 * kWSlice;     // 65792 floats (257 KB)
constexpr int kABuf    = 16 * kAStride;    // 2112 floats per A buffer
constexpr size_t kLdsBytes = (size_t)(kSBFloats + 2 * kABuf) * 4;  // 280064 B < 320 KB

// ---- CDNA5 async global->LDS DMA (ASYNCcnt path), via inline asm ------------
__device__ __forceinline__ uint32_t lds_u32(const void* p) {
  // AS3 pointer value == byte offset within the workgroup's LDS window
  return (uint32_t)(uintptr_t)(__attribute__((address_space(3))) const void*)p;
}
__device__ __forceinline__ void async_b128(uint32_t lds, const float* g) {
  asm volatile("global_load_async_to_lds_b128 %0, %1, off"
               :: "v"(lds), "v"(g) : "memory");
}
__device__ __forceinline__ void wait_async0() {
  asm volatile("s_wait_asynccnt 0x0" ::: "memory");
}

// ---------------------------------------------------------------------------
// Kernel 0: zero per-step barrier counters (ws state must be re-inited per call)
// ---------------------------------------------------------------------------
__global__ void zero_counters(unsigned int* __restrict__ c, int n) {
  int i = blockIdx.x * blockDim.x + threadIdx.x;
  if (i < n) c[i] = 0u;
}

// ---------------------------------------------------------------------------
// Kernel 1: xi = x @ Wi^T + bi  -> d_out[0 .. T*B*H)   (memory-bound)
// Dual accumulators break the WMMA D->C dependence chain.
// ---------------------------------------------------------------------------
__global__ __launch_bounds__(256) void xi_gemm(
    const float* __restrict__ x, const float* __restrict__ Wi,
    const float* __restrict__ bi, float* __restrict__ out)
{
  const int wave = threadIdx.x >> 5;
  const int lane = threadIdx.x & 31;
  const int tile = blockIdx.x * 8 + wave;
  const int NT   = kH / 16;
  const int mt   = tile / NT;
  const int nt   = tile % NT;
  const int r    = lane & 15;
  const int ko   = (lane >> 4) * 2;

  const float* arow = x  + (size_t)(mt * 16 + r) * kI;
  const float* brow = Wi + (size_t)(nt * 16 + r) * kI;

  v8f c0 = {}, c1 = {};
  for (int k0 = 0; k0 < kI; k0 += 8) {
    v2f a0 = *(const v2f*)(arow + k0 + ko);
    v2f b0 = *(const v2f*)(brow + k0 + ko);
    c0 = __builtin_amdgcn_wmma_f32_16x16x4_f32(false, a0, false, b0,
                                               (short)0, c0, false, false);
    v2f a1 = *(const v2f*)(arow + k0 + 4 + ko);
    v2f b1 = *(const v2f*)(brow + k0 + 4 + ko);
    c1 = __builtin_amdgcn_wmma_f32_16x16x4_f32(false, a1, false, b1,
                                               (short)0, c1, false, false);
  }
  const v8f c = c0 + c1;
  const int n     = nt * 16 + r;
  const int mbase = mt * 16 + (lane >> 4) * 8;
  const float bias = bi[n];
  #pragma unroll
  for (int j = 0; j < 8; ++j)
    out[(size_t)(mbase + j) * kH + n] = c[j] + bias;
}

// ---------------------------------------------------------------------------
// Kernel 2: persistent scan. Wh LDS-resident (256 KB/WG); h-strip async-staged
// and double-buffered; fp32 WMMA with dual accumulators; per-step grid barrier.
// ---------------------------------------------------------------------------
__global__ __launch_bounds__(128) void ctrnn_scan(
    const float* __restrict__ h0, const float* __restrict__ Wh,
    const float* __restrict__ bh, float* __restrict__ out,
    unsigned int* __restrict__ counters)
{
  extern __shared__ float smem[];
  float* sB = smem;                    // [4 waves][16][kBStride]
  float* sA = smem + kSBFloats;        // [2 bufs][16][kAStride]

  const int tid    = threadIdx.x;
  const int wave   = tid >> 5;
  const int lane   = tid & 31;
  const int mstrip = blockIdx.x >> 4;               // 0..7
  const int ntile  = (blockIdx.x & 15) * 4 + wave;  // 0..63
  const int r      = lane & 15;
  const int ko     = (lane >> 4) * 2;
  const int n      = ntile * 16 + r;
  const float bias = bh[n];
  float* hfinal    = out + (size_t)kT * kB * kH;
  float* sBw       = sB + wave * kWSlice;

  // ---- one-time: DMA this wave's 16 Wh rows (64 KB) into LDS --------------
  {
    const float* wbase = Wh + (size_t)(ntile * 16) * kH + lane * 4;
    uint32_t dbase = lds_u32(sBw + lane * 4);
    for (int row = 0; row < 16; ++row) {
      const float* src = wbase + (size_t)row * kH;
      uint32_t dst = dbase + (uint32_t)(row * kBStride) * 4u;
      #pragma unroll
      for (int i = 0; i < 8; ++i)
        async_b128(dst + i * 512u, src + i * 128);   // 512B chunks, coalesced
    }
  }
  wait_async0();
  __syncthreads();

  const int mbase = mstrip * 16 + (lane >> 4) * 8;
  const int arow  = tid >> 3;          // 0..15 (A-strip cooperative load)
  const int acol  = (tid & 7) * 16;    // 0..112

  for (int t = 0; t < kT; ++t) {
    const float* hp = t ? out + (size_t)(t - 1) * kB * kH : h0;
    float*       ot = out + (size_t)t * kB * kH;   // holds xi[t], becomes h[t]

    // prefetch elementwise operands early (hidden under the GEMM)
    float xi_r[8], hp_r[8];
    #pragma unroll
    for (int j = 0; j < 8; ++j) {
      const size_t idx = (size_t)(mbase + j) * kH + n;
      xi_r[j] = ot[idx];
      hp_r[j] = hp[idx];
    }

    // prologue: stage K-block 0 of the h strip
    {
      const float* src = hp + (size_t)(mstrip * 16 + arow) * kH + acol;
      uint32_t dst = lds_u32(sA + arow * kAStride + acol);
      #pragma unroll
      for (int i = 0; i < 4; ++i) async_b128(dst + i * 16u, src + i * 4);
    }

    v8f c0 = {}, c1 = {};
    for (int ib = 0; ib < kH / kBK; ++ib) {
      wait_async0();        // my block-ib DMAs landed
      __syncthreads();      // everyone's landed; buffer (ib+1)&1 free to refill
      if (ib + 1 < kH / kBK) {
        const float* src = hp + (size_t)(mstrip * 16 + arow) * kH
                              + (ib + 1) * kBK + acol;
        uint32_t dst = lds_u32(sA + ((ib + 1) & 1) * kABuf
                               + arow * kAStride + acol);
        #pragma unroll
        for (int i = 0; i < 4; ++i) async_b128(dst + i * 16u, src + i * 4);
      }
      const float* Ap = sA + (ib & 1) * kABuf + r * kAStride;
      const float* Bp = sBw + r * kBStride + ib * kBK;
      #pragma unroll
      for (int kk = 0; kk < kBK; kk += 8) {
        v2f a0 = *(const v2f*)(Ap + kk + ko);
        v2f b0 = *(const v2f*)(Bp + kk + ko);
        c0 = __builtin_amdgcn_wmma_f32_16x16x4_f32(false, a0, false, b0,
                                                   (short)0, c0, false, false);
        v2f a1 = *(const v2f*)(Ap + kk + 4 + ko);
        v2f b1 = *(const v2f*)(Bp + kk + 4 + ko);
        c1 = __builtin_amdgcn_wmma_f32_16x16x4_f32(false, a1, false, b1,
                                                   (short)0, c1, false, false);
      }
    }
    const v8f c = c0 + c1;

    // elementwise: relu + leaky integration; overwrite xi[t] with h[t]
    #pragma unroll
    for (int j = 0; j < 8; ++j) {
      const size_t idx = (size_t)(mbase + j) * kH + n;
      float hnew = xi_r[j] + c[j] + bias;
      hnew = fmaxf(hnew, 0.0f);
      const float hcur = hp_r[j] * (1.0f - kAlpha) + hnew * kAlpha;
      ot[idx] = hcur;
      if (t == kT - 1) hfinal[idx] = hcur;
    }

    // ---- device-wide barrier for step t ----
    __threadfence();          // release this thread's h[t] stores to device scope
    __syncthreads();
    if (tid == 0) {
      __hip_atomic_fetch_add(&counters[t], 1u, __ATOMIC_RELEASE,
                             __HIP_MEMORY_SCOPE_AGENT);
      while (__hip_atomic_load(&counters[t], __ATOMIC_RELAXED,
                               __HIP_MEMORY_SCOPE_AGENT) < (unsigned)kNWG) {}
    }
    __syncthreads();
    // every wave (both CUs of the WGP) must invalidate L0 before reading h[t]
    __builtin_amdgcn_fence(__ATOMIC_ACQUIRE, "agent");
  }
}

// ---------------------------------------------------------------------------
extern "C" void kernel_launch(void* const* d_in, const int* in_sizes, int n_in,
                              void* d_out, int out_size, void* d_ws, size_t ws_size,
                              hipStream_t stream) {
  const float* x  = (const float*)d_in[0];
  const float* h0 = (const float*)d_in[1];
  const float* Wi = (const float*)d_in[2];
  const float* bi = (const float*)d_in[3];
  const float* Wh = (const float*)d_in[4];
  const float* bh = (const float*)d_in[5];
  float* out = (float*)d_out;
  unsigned int* counters = (unsigned int*)d_ws;   // kT uints (2 KB)

  zero_counters<<<2, 256, 0, stream>>>(counters, kT);

  const int xi_tiles  = (kT * kB / 16) * (kH / 16);
  const int xi_blocks = xi_tiles / 8;
  xi_gemm<<<xi_blocks, 256, 0, stream>>>(x, Wi, bi, out);

  ctrnn_scan<<<kNWG, 128, 0, stream>>>(h0, Wh, bh, out, counters);
}